// VectorQuantizer_78280073937157
// MI455X (gfx1250) — compile-verified
//
#include <hip/hip_runtime.h>
#include <hip/hip_bf16.h>
#include <math.h>

typedef float v2f __attribute__((ext_vector_type(2)));
typedef float v8f __attribute__((ext_vector_type(8)));

// Problem constants
static constexpr int    VQ_N    = 65536;        // 32 * 2048 rows
static constexpr int    VQ_D    = 64;           // e_dim
static constexpr int    VQ_NE   = 1024;         // codebook entries
static constexpr int    VQ_T    = 2048;         // time dim
static constexpr float  VQ_BETA = 0.25f;

// d_out layout (float element offsets), outputs concatenated in return order:
// z_q[32,64,2048], loss, perplexity, min_encodings[65536,1024], indices[65536,1]
static constexpr size_t OFF_ZQ   = 0;
static constexpr size_t OFF_LOSS = 4194304;     // 32*64*2048
static constexpr size_t OFF_PERP = 4194305;
static constexpr size_t OFF_ENC  = 4194306;
static constexpr size_t OFF_IDX  = 71303170;    // OFF_ENC + 65536*1024

// workspace layout (float element offsets)
static constexpr int WS_ENORM = 0;              // 1024 floats: ||e_j||^2
static constexpr int WS_CNT   = 1024;           // 1024 floats: histogram
static constexpr int WS_LOSS  = 2048;           // 1 float: sum of (zq - z)^2

// ---------------------------------------------------------------------------
// K0: per-code squared norms + zero accumulators
// ---------------------------------------------------------------------------
__global__ void vq_init(const float* __restrict__ emb, float* __restrict__ ws) {
    int i = blockIdx.x * blockDim.x + threadIdx.x;
    if (i < VQ_NE) {
        const float* e = emb + (size_t)i * VQ_D;
        float s = 0.f;
        #pragma unroll
        for (int k = 0; k < VQ_D; ++k) s += e[k] * e[k];
        ws[WS_ENORM + i] = s;
        ws[WS_CNT + i]   = 0.f;
        if (i == 0) ws[WS_LOSS] = 0.f;
    }
}

// ---------------------------------------------------------------------------
// K1: zero the one-hot region (67,108,864 floats). Base is 8B-aligned -> b64.
// grid 32768 x 256, 4 float2 per thread (coalesced)
// ---------------------------------------------------------------------------
__global__ void vq_zero_enc(float* __restrict__ out) {
    v2f* p = (v2f*)(out + OFF_ENC);
    size_t base = (size_t)blockIdx.x * 1024 + threadIdx.x;
    #pragma unroll
    for (int k = 0; k < 4; ++k) {
        v2f zero = {0.f, 0.f};
        p[base + (size_t)k * 256] = zero;
    }
}

// ---------------------------------------------------------------------------
// K2: main VQ kernel. One wave32 per 16-row tile; WMMA f32 16x16x4 GEMM
// Z(16x64) x E_chunk^T(64x16), fused argmin + all output writes.
// ---------------------------------------------------------------------------
__global__ void __launch_bounds__(128)
vq_main(const float* __restrict__ z, const float* __restrict__ emb,
        float* ws, float* out) {
    const int lane = threadIdx.x & 31;
    const int wave = threadIdx.x >> 5;
    const int row0 = (blockIdx.x * 4 + wave) * 16;   // 16-row tile base
    const int m    = lane & 15;                       // A row / B col within tile
    const int kh   = lane >> 4;                       // K-half select

    float minv[8];
    int   mini[8];
    #pragma unroll
    for (int r = 0; r < 8; ++r) { minv[r] = 3.4e38f; mini[r] = 0; }

    const float* enorm = ws + WS_ENORM;

#if __has_builtin(__builtin_amdgcn_wmma_f32_16x16x4_f32)
    // A fragments: 16x4 f32 layout -> lane m holds Z[row0+m][4t+2kh .. +1]
    const float* zrow = z + (size_t)(row0 + m) * VQ_D;
    v2f a[16];
    #pragma unroll
    for (int t = 0; t < 16; ++t)
        a[t] = *(const v2f*)(zrow + 4 * t + 2 * kh);

    for (int c = 0; c < VQ_NE / 16; ++c) {
        const int j = c * 16 + m;                     // code index for this lane
        const float* erow = emb + (size_t)j * VQ_D;
        v8f acc = {0.f, 0.f, 0.f, 0.f, 0.f, 0.f, 0.f, 0.f};
        #pragma unroll
        for (int t = 0; t < 16; ++t) {
            // B fragment: 4x16 layout mirrors A -> contiguous float2 of E row j
            v2f b = *(const v2f*)(erow + 4 * t + 2 * kh);
            acc = __builtin_amdgcn_wmma_f32_16x16x4_f32(
                false, a[t], false, b, (short)0, acc, false, false);
        }
        const float en = enorm[j];
        #pragma unroll
        for (int r = 0; r < 8; ++r) {                 // reg r <-> row r + 8*kh
            float d = en - 2.0f * acc[r];             // ||e||^2 - 2 z.e
            if (d < minv[r]) { minv[r] = d; mini[r] = j; }
        }
    }
#else
    // Scalar fallback (same register<->row mapping as WMMA path)
    for (int c = 0; c < VQ_NE / 16; ++c) {
        const int j = c * 16 + m;
        const float* erow = emb + (size_t)j * VQ_D;
        const float en = enorm[j];
        #pragma unroll
        for (int r = 0; r < 8; ++r) {
            const float* zr = z + (size_t)(row0 + r + 8 * kh) * VQ_D;
            float dot = 0.f;
            for (int k = 0; k < VQ_D; ++k) dot += zr[k] * erow[k];
            float d = en - 2.0f * dot;
            if (d < minv[r]) { minv[r] = d; mini[r] = j; }
        }
    }
#endif

    // Min-reduce across the 16 lanes of each half (xor 1,2,4,8 stays in half).
    // Tie-break on lower index to match jnp.argmin.
    #pragma unroll
    for (int off = 1; off <= 8; off <<= 1) {
        #pragma unroll
        for (int r = 0; r < 8; ++r) {
            float ov = __shfl_xor(minv[r], off, 32);
            int   oi = __shfl_xor(mini[r], off, 32);
            if (ov < minv[r] || (ov == minv[r] && oi < mini[r])) {
                minv[r] = ov; mini[r] = oi;
            }
        }
    }

    float* out_idx = out + OFF_IDX;
    float* out_enc = out + OFF_ENC;
    if (m == 0) {                     // lane 0 -> rows 0..7, lane 16 -> rows 8..15
        #pragma unroll
        for (int r = 0; r < 8; ++r) {
            int row = row0 + r + 8 * kh;
            int idx = mini[r];
            out_idx[row] = (float)idx;                          // int->float concat
            out_enc[(size_t)row * VQ_NE + idx] = 1.0f;          // one-hot
            atomicAdd(&ws[WS_CNT + idx], 1.0f);                 // histogram
        }
    }

    // z_q gather + transpose to [B, D, T], plus loss partial sum
    float* out_zq = out + OFF_ZQ;
    float lsum = 0.f;
    #pragma unroll
    for (int mm = 0; mm < 16; ++mm) {
        int idx = __shfl(mini[mm & 7], (mm >> 3) << 4, 32);     // broadcast row idx
        int row = row0 + mm;
        int bb  = row >> 11;                                    // / 2048
        int tt  = row & 2047;
        const float* er = emb + (size_t)idx * VQ_D;
        const float* zr = z + (size_t)row * VQ_D;
        float e0 = er[lane],      e1 = er[lane + 32];
        float z0 = zr[lane],      z1 = zr[lane + 32];
        out_zq[((size_t)bb * VQ_D + lane)      * VQ_T + tt] = e0;
        out_zq[((size_t)bb * VQ_D + lane + 32) * VQ_T + tt] = e1;
        float d0 = e0 - z0, d1 = e1 - z1;
        lsum += d0 * d0 + d1 * d1;
    }
    #pragma unroll
    for (int off = 16; off > 0; off >>= 1) lsum += __shfl_xor(lsum, off, 32);
    if (lane == 0) atomicAdd(&ws[WS_LOSS], lsum);
}

// ---------------------------------------------------------------------------
// K3: finalize loss + perplexity
// ---------------------------------------------------------------------------
__global__ void vq_final(const float* __restrict__ ws, float* __restrict__ out) {
    __shared__ float sh[256];
    float s = 0.f;
    for (int i = threadIdx.x; i < VQ_NE; i += 256) {
        float p = ws[WS_CNT + i] * (1.0f / (float)VQ_N);
        s += p * logf(p + 1e-10f);
    }
    sh[threadIdx.x] = s;
    __syncthreads();
    for (int o = 128; o > 0; o >>= 1) {
        if (threadIdx.x < o) sh[threadIdx.x] += sh[threadIdx.x + o];
        __syncthreads();
    }
    if (threadIdx.x == 0) {
        out[OFF_PERP] = expf(-sh[0]);
        // both MSE terms are numerically identical -> (1 + beta) * mean
        out[OFF_LOSS] = (1.0f + VQ_BETA) * ws[WS_LOSS]
                        * (1.0f / (float)(VQ_N * VQ_D));
    }
}

// ---------------------------------------------------------------------------
extern "C" void kernel_launch(void* const* d_in, const int* in_sizes, int n_in,
                              void* d_out, int out_size, void* d_ws, size_t ws_size,
                              hipStream_t stream) {
    (void)in_sizes; (void)n_in; (void)out_size; (void)ws_size;
    const float* z   = (const float*)d_in[0];
    const float* emb = (const float*)d_in[1];
    float* out = (float*)d_out;
    float* ws  = (float*)d_ws;

    vq_init<<<4, 256, 0, stream>>>(emb, ws);
    vq_zero_enc<<<32768, 256, 0, stream>>>(out);
    vq_main<<<VQ_N / 64, 128, 0, stream>>>(z, emb, ws, out);
    vq_final<<<1, 256, 0, stream>>>(ws, out);
}